// MPNN_22952305230124
// MI455X (gfx1250) — compile-verified
//
#include <hip/hip_runtime.h>

// ---------------------------------------------------------------------------
// MPNN (4-step message passing GNN) for MI455X / gfx1250.
//  - fp32 WMMA (v_wmma_f32_16x16x4_f32) for all GEMMs: problem is L2
//    gather/atomic bound (working set < 192MB L2), so fp32 precision is free
//    and the matrix pipe offloads GEMMs from the VALU.
//  - CDNA5 async global->LDS loads (ASYNCcnt) for the random neighbor gather
//    (no VGPR round-trip; all 32 gathers per lane in flight at once) and for
//    the contiguous GRU tile staging (B128).
// ---------------------------------------------------------------------------

#define FDIM   32            // F
#define WAVES  8             // waves per block (256 threads, wave32)

typedef __attribute__((ext_vector_type(2))) float v2f;
typedef __attribute__((ext_vector_type(8))) float v8f;

__device__ __forceinline__ v8f wmma4(v2f a, v2f b, v8f c) {
  // (neg_a, A, neg_b, B, c_mod, C, reuse_a, reuse_b)
  return __builtin_amdgcn_wmma_f32_16x16x4_f32(false, a, false, b, (short)0, c,
                                               false, false);
}

__device__ __forceinline__ float sigmoidf_(float x) {
  return 1.0f / (1.0f + __expf(-x));
}

// async global->LDS: deposits directly into LDS, tracked by ASYNCcnt.
__device__ __forceinline__ void async_ld_b32(unsigned lds_off, const void* g) {
  asm volatile("global_load_async_to_lds_b32 %0, %1, off"
               :: "v"(lds_off), "v"(g) : "memory");
}
__device__ __forceinline__ void async_ld_b128(unsigned lds_off, const void* g) {
  asm volatile("global_load_async_to_lds_b128 %0, %1, off"
               :: "v"(lds_off), "v"(g) : "memory");
}
__device__ __forceinline__ void wait_async0() {
  asm volatile("s_wait_asynccnt 0x0" ::: "memory");
}
__device__ __forceinline__ unsigned lds_off32(const void* p) {
  // low 32 bits of a generic pointer into __shared__ = LDS byte offset
  return (unsigned)(size_t)p;
}

// ---------------------------------------------------------------------------
// zero a float buffer (grid-stride)
// ---------------------------------------------------------------------------
__global__ void zero_kernel(float* __restrict__ p, long long n) {
  long long i = (long long)blockIdx.x * blockDim.x + threadIdx.x;
  long long stride = (long long)gridDim.x * blockDim.x;
  for (; i < n; i += stride) p[i] = 0.0f;
}

// ---------------------------------------------------------------------------
// Message kernel: per 16-edge tile (one wave):
//   neighbor(16x64) = gather(state, pair)    (async global->LDS, ASYNCcnt)
//   h1 = relu(neighbor @ w1^T + b1)  (16x32) -> 2 tiles x 16 WMMA
//   m  = relu(h1 @ w2^T + b2)        (16x32) -> 2 tiles x  8 WMMA
//   agg[pair[2e, n], n] += m[e, n]           (atomic scatter)
// ---------------------------------------------------------------------------
__global__ void msg_kernel(const float* __restrict__ state,
                           const int*   __restrict__ pair,
                           const float* __restrict__ w1,   // (32,64) row-major
                           const float* __restrict__ b1,   // (32)
                           const float* __restrict__ w2,   // (32,32)
                           const float* __restrict__ b2,   // (32)
                           float* __restrict__ agg,
                           int E) {
  __shared__ float s_w1[32 * 64];
  __shared__ float s_w2[32 * 32];
  __shared__ float s_b1[32];
  __shared__ float s_b2[32];
  __shared__ float s_nb[WAVES * 16 * 64];   // gathered neighbor tiles
  __shared__ float s_h1[WAVES * 16 * 32];   // layer-1 activations

  const int tid  = threadIdx.x;
  const int wv   = tid >> 5;
  const int lane = tid & 31;
  const int hi   = lane >> 4;     // lane half: selects K offset / row offset
  const int lr   = lane & 15;

  // stage weights (block cooperative)
  for (int i = tid; i < 32 * 64; i += blockDim.x) s_w1[i] = w1[i];
  for (int i = tid; i < 32 * 32; i += blockDim.x) s_w2[i] = w2[i];
  if (tid < 32) { s_b1[tid] = b1[tid]; s_b2[tid] = b2[tid]; }

  // gather 16 edges x 64 features into this wave's LDS tile.
  // All 32 random gathers per lane issued async; one wait at the end.
  const int e0 = (blockIdx.x * WAVES + wv) * 16;
  float* nb = s_nb + wv * 16 * 64;
  #pragma unroll
  for (int i = 0; i < 32; ++i) {
    int id = lane + 32 * i;            // 0..1023
    int er = id >> 6;                  // edge row within tile
    int j  = id & 63;                  // neighbor feature 0..63
    int e  = e0 + er;
    if (e < E) {
      int prow = (j < 32) ? (2 * e) : (2 * e + 1);
      int col  = j & 31;
      int idx  = pair[prow * FDIM + col];
      async_ld_b32(lds_off32(&nb[er * 64 + j]),
                   &state[(long long)idx * FDIM + col]);
    } else {
      nb[er * 64 + j] = 0.0f;          // disjoint addresses: no hazard
    }
  }
  wait_async0();
  __syncthreads();

  // ---- layer 1: (16x64) @ (64x32) ----
  float* h1 = s_h1 + wv * 16 * 32;
  #pragma unroll
  for (int tn = 0; tn < 2; ++tn) {
    v8f c = {};
    #pragma unroll
    for (int k0 = 0; k0 < 64; k0 += 4) {
      int k = k0 + 2 * hi;
      v2f a; a.x = nb[lr * 64 + k];         a.y = nb[lr * 64 + k + 1];
      int n = tn * 16 + lr;                  // output column
      v2f b; b.x = s_w1[n * 64 + k];        b.y = s_w1[n * 64 + k + 1];
      c = wmma4(a, b, c);
    }
    int n = tn * 16 + lr;
    float bias = s_b1[n];
    #pragma unroll
    for (int v = 0; v < 8; ++v) {
      float x = c[v] + bias;
      h1[(v + 8 * hi) * 32 + n] = x > 0.0f ? x : 0.0f;
    }
  }
  __syncthreads();

  // ---- layer 2: (16x32) @ (32x32), fused bias+relu+atomic scatter ----
  #pragma unroll
  for (int tn = 0; tn < 2; ++tn) {
    v8f c = {};
    #pragma unroll
    for (int k0 = 0; k0 < 32; k0 += 4) {
      int k = k0 + 2 * hi;
      v2f a; a.x = h1[lr * 32 + k];         a.y = h1[lr * 32 + k + 1];
      int n = tn * 16 + lr;
      v2f b; b.x = s_w2[n * 32 + k];        b.y = s_w2[n * 32 + k + 1];
      c = wmma4(a, b, c);
    }
    int n = tn * 16 + lr;
    float bias = s_b2[n];
    #pragma unroll
    for (int v = 0; v < 8; ++v) {
      int row = v + 8 * hi;
      int e = e0 + row;
      if (e < E) {
        float x = c[v] + bias;
        x = x > 0.0f ? x : 0.0f;
        int idx = pair[(2 * e) * FDIM + n];
        atomicAdd(&agg[(long long)idx * FDIM + n], x);
      }
    }
  }
}

// ---------------------------------------------------------------------------
// GRU kernel: per 16-node tile (one wave):
//   stage agg/state tiles via async B128 global->LDS
//   gi = agg   @ w_ih^T + b_ih   (16x96)  -> 6 tiles x 8 WMMA
//   gh = state @ w_hh^T + b_hh   (16x96)  -> 6 tiles x 8 WMMA
//   gates + new state, all lane-local (tiles tn, tn+2, tn+4 share positions)
// ---------------------------------------------------------------------------
__global__ void gru_kernel(const float* __restrict__ state,
                           const float* __restrict__ agg,
                           const float* __restrict__ w_ih,  // (96,32)
                           const float* __restrict__ b_ih,  // (96)
                           const float* __restrict__ w_hh,  // (96,32)
                           const float* __restrict__ b_hh,  // (96)
                           float* __restrict__ new_state,
                           int N) {
  __shared__ float s_wih[96 * 32];
  __shared__ float s_whh[96 * 32];
  __shared__ float s_bih[96];
  __shared__ float s_bhh[96];
  __shared__ float s_a[WAVES * 16 * 32];
  __shared__ float s_s[WAVES * 16 * 32];

  const int tid  = threadIdx.x;
  const int wv   = tid >> 5;
  const int lane = tid & 31;
  const int hi   = lane >> 4;
  const int lr   = lane & 15;

  for (int i = tid; i < 96 * 32; i += blockDim.x) {
    s_wih[i] = w_ih[i];
    s_whh[i] = w_hh[i];
  }
  if (tid < 96) { s_bih[tid] = b_ih[tid]; s_bhh[tid] = b_hh[tid]; }

  const int n0 = (blockIdx.x * WAVES + wv) * 16;
  float* ta = s_a + wv * 16 * 32;
  float* ts = s_s + wv * 16 * 32;
  // 16x32 tile per array = 128 x B128 transfers per wave = 4 per lane each
  #pragma unroll
  for (int i = 0; i < 4; ++i) {
    int id  = lane + 32 * i;       // 0..127
    int row = id >> 3;             // 0..15
    int c4  = (id & 7) * 4;        // 0,4,...,28
    int nd  = n0 + row;
    if (nd < N) {
      async_ld_b128(lds_off32(&ta[row * 32 + c4]),
                    &agg  [(long long)nd * FDIM + c4]);
      async_ld_b128(lds_off32(&ts[row * 32 + c4]),
                    &state[(long long)nd * FDIM + c4]);
    } else {
      #pragma unroll
      for (int q = 0; q < 4; ++q) {
        ta[row * 32 + c4 + q] = 0.0f;
        ts[row * 32 + c4 + q] = 0.0f;
      }
    }
  }
  wait_async0();
  __syncthreads();

  v8f gi_t[6], gh_t[6];
  #pragma unroll
  for (int tn = 0; tn < 6; ++tn) {
    v8f ci = {}, ch = {};
    #pragma unroll
    for (int k0 = 0; k0 < 32; k0 += 4) {
      int k = k0 + 2 * hi;
      int n = tn * 16 + lr;                  // gate column 0..95
      v2f ai; ai.x = ta[lr * 32 + k];       ai.y = ta[lr * 32 + k + 1];
      v2f ah; ah.x = ts[lr * 32 + k];       ah.y = ts[lr * 32 + k + 1];
      v2f bi; bi.x = s_wih[n * 32 + k];     bi.y = s_wih[n * 32 + k + 1];
      v2f bh; bh.x = s_whh[n * 32 + k];     bh.y = s_whh[n * 32 + k + 1];
      ci = wmma4(ai, bi, ci);
      ch = wmma4(ah, bh, ch);
    }
    int n = tn * 16 + lr;
    float bi = s_bih[n], bh = s_bhh[n];
    #pragma unroll
    for (int v = 0; v < 8; ++v) { ci[v] += bi; ch[v] += bh; }
    gi_t[tn] = ci; gh_t[tn] = ch;
  }

  // gates: f = tn*16 + lr (tn=0,1); r->tiles tn, z->tn+2, n->tn+4
  #pragma unroll
  for (int tn = 0; tn < 2; ++tn) {
    int f = tn * 16 + lr;
    #pragma unroll
    for (int v = 0; v < 8; ++v) {
      int row = v + 8 * hi;
      int nd  = n0 + row;
      float r  = sigmoidf_(gi_t[tn    ][v] + gh_t[tn    ][v]);
      float z  = sigmoidf_(gi_t[tn + 2][v] + gh_t[tn + 2][v]);
      float nn = tanhf    (gi_t[tn + 4][v] + r * gh_t[tn + 4][v]);
      float h_old = ts[row * 32 + f];
      float out = (1.0f - z) * nn + z * h_old;
      if (nd < N) new_state[(long long)nd * FDIM + f] = out;
    }
  }
}

// ---------------------------------------------------------------------------
// column sum over all nodes: feature[c] = sum_n state[n, c]
// ---------------------------------------------------------------------------
__global__ void colsum_kernel(const float* __restrict__ state,
                              float* __restrict__ feature, int N) {
  long long tid    = (long long)blockIdx.x * blockDim.x + threadIdx.x;
  long long stride = (long long)gridDim.x * blockDim.x;   // multiple of 32
  int col = (int)(tid & 31);
  float s = 0.0f;
  long long total = (long long)N * FDIM;
  for (long long i = tid; i < total; i += stride) s += state[i];
  atomicAdd(&feature[col], s);
}

// ---------------------------------------------------------------------------
// readout MLP: 32 -> 64 -> 64 -> 1 (single block, 64 threads)
// ---------------------------------------------------------------------------
__global__ void readout_kernel(const float* __restrict__ feature,
                               const float* __restrict__ w1,  // (64,32)
                               const float* __restrict__ b1,  // (64)
                               const float* __restrict__ w2,  // (64,64)
                               const float* __restrict__ b2,  // (64)
                               const float* __restrict__ w3,  // (1,64)
                               const float* __restrict__ b3,  // (1)
                               float* __restrict__ out) {
  __shared__ float s_f[32];
  __shared__ float s_h1[64];
  __shared__ float s_h2[64];
  int t = threadIdx.x;
  if (t < 32) s_f[t] = feature[t];
  __syncthreads();
  float acc = 0.0f;
  #pragma unroll
  for (int k = 0; k < 32; ++k) acc += w1[t * 32 + k] * s_f[k];
  s_h1[t] = fmaxf(acc + b1[t], 0.0f);
  __syncthreads();
  acc = 0.0f;
  #pragma unroll
  for (int k = 0; k < 64; ++k) acc += w2[t * 64 + k] * s_h1[k];
  s_h2[t] = fmaxf(acc + b2[t], 0.0f);
  __syncthreads();
  if (t == 0) {
    float o = 0.0f;
    for (int k = 0; k < 64; ++k) o += w3[k] * s_h2[k];
    out[0] = o + b3[0];
  }
}

// ---------------------------------------------------------------------------
extern "C" void kernel_launch(void* const* d_in, const int* in_sizes, int n_in,
                              void* d_out, int out_size, void* d_ws, size_t ws_size,
                              hipStream_t stream) {
  const float* link_state = (const float*)d_in[0];
  const int*   pair       = (const int*)  d_in[1];
  const float* msg_w1     = (const float*)d_in[2];
  const float* msg_b1     = (const float*)d_in[3];
  const float* msg_w2     = (const float*)d_in[4];
  const float* msg_b2     = (const float*)d_in[5];
  const float* gru_w_ih   = (const float*)d_in[6];
  const float* gru_b_ih   = (const float*)d_in[7];
  const float* gru_w_hh   = (const float*)d_in[8];
  const float* gru_b_hh   = (const float*)d_in[9];
  const float* ro_w1      = (const float*)d_in[10];
  const float* ro_b1      = (const float*)d_in[11];
  const float* ro_w2      = (const float*)d_in[12];
  const float* ro_b2      = (const float*)d_in[13];
  const float* ro_w3      = (const float*)d_in[14];
  const float* ro_b3      = (const float*)d_in[15];

  const int N = in_sizes[0] / FDIM;        // 100000
  const int E = in_sizes[1] / (2 * FDIM);  // 500000

  float* ws      = (float*)d_ws;
  float* stA     = ws;
  float* stB     = ws + (size_t)N * FDIM;
  float* agg     = ws + 2 * (size_t)N * FDIM;
  float* feature = ws + 3 * (size_t)N * FDIM;

  const long long aggN = (long long)N * FDIM;
  const int zeroBlocks = (int)((aggN + 255) / 256);
  const int msgBlocks  = (E + 16 * WAVES - 1) / (16 * WAVES);
  const int gruBlocks  = (N + 16 * WAVES - 1) / (16 * WAVES);

  const float* cur = link_state;
  float* bufs[2] = {stA, stB};
  for (int t = 0; t < 4; ++t) {
    zero_kernel<<<zeroBlocks, 256, 0, stream>>>(agg, aggN);
    msg_kernel<<<msgBlocks, 256, 0, stream>>>(cur, pair, msg_w1, msg_b1,
                                              msg_w2, msg_b2, agg, E);
    float* nxt = bufs[t & 1];
    gru_kernel<<<gruBlocks, 256, 0, stream>>>(cur, agg, gru_w_ih, gru_b_ih,
                                              gru_w_hh, gru_b_hh, nxt, N);
    cur = nxt;
  }

  zero_kernel<<<1, 64, 0, stream>>>(feature, FDIM);
  colsum_kernel<<<1024, 256, 0, stream>>>(cur, feature, N);
  readout_kernel<<<1, 64, 0, stream>>>(feature, ro_w1, ro_b1, ro_w2, ro_b2,
                                       ro_w3, ro_b3, (float*)d_out);
}